// MaskedGeneralInputProcessor_2207613190214
// MI455X (gfx1250) — compile-verified
//
#include <hip/hip_runtime.h>

// MaskedGeneralInputProcessor for MI455X (gfx1250, wave32).
//
// out[b,t,e] = (sum_{d < asset_dims[b]+1} x[b,t,d]) / (asset_dims[b]+1) * w[e] + b[e]
//
// Store-bandwidth bound: 128 MiB out / 8 MiB in => ~6 us roofline at 23.3 TB/s.
// The E-broadcast is done as a rank-1 WMMA (V_WMMA_F32_16X16X4_F32):
//   A[16x4]  : column K=0 = mean of 16 t-rows, K=1..3 = 0
//   B[4x16]  : row    K=0 = w[e0..e0+15],      K=1..3 = 0
//   C[16x16] : bias replicated down rows
// One WMMA -> 256 f32 outputs, minimizing instructions per stored byte.

typedef __attribute__((ext_vector_type(2))) float v2f;
typedef __attribute__((ext_vector_type(8))) float v8f;

#define B_ 64
#define T_ 2048
#define D_ 16
#define E_ 256

__global__ __launch_bounds__(256) void
masked_proj_wmma_kernel(const float* __restrict__ x,
                        const int* __restrict__ asset_dims,
                        const float* __restrict__ w,
                        const float* __restrict__ bias,
                        float* __restrict__ out) {
    constexpr int TILES_PER_B = T_ / 16;              // 128 t-tiles per batch row

    const int blk  = blockIdx.x;
    const int bIdx = blk / TILES_PER_B;               // batch index
    const int t0   = (blk % TILES_PER_B) * 16;        // first t-row of this tile

    const int lane = threadIdx.x & 31;                // wave32
    const int wave = threadIdx.x >> 5;                // 8 waves per block
    const int half = lane >> 4;                       // 0: lanes 0-15, 1: lanes 16-31
    const int l16  = lane & 15;

    // Uniform per-block scalar: number of real features (1..16).
    const int   n_real = asset_dims[bIdx] + 1;
    const float inv_n  = 1.0f / (float)n_real;

    // ---- masked mean over D=16 for row (t0 + l16) ----
    // Only lanes 0-15 feed the A-matrix; redundant reads across waves hit L0/L2
    // (x is 8 MiB, far under the 192 MB L2) and cost nothing vs the store stream.
    const float* __restrict__ row = x + ((size_t)bIdx * T_ + (size_t)(t0 + l16)) * D_;
    float s = 0.0f;
#pragma unroll
    for (int d = 0; d < D_; ++d) {
        float v = row[d];
        s += (d < n_real) ? v : 0.0f;                 // v_cndmask, no divergence
    }
    const float mean_val = s * inv_n;

    // ---- A-matrix (16x4 f32): lanes 0-15 -> M rows, VGPR0 = K0, VGPR1 = K1;
    //      lanes 16-31 -> K2/K3. Only K=0 is nonzero (rank-1 product).
    v2f A;
    A.x = (half == 0) ? mean_val : 0.0f;
    A.y = 0.0f;

    // Each wave covers two adjacent 16-wide e-tiles => 128 contiguous bytes/row.
#pragma unroll
    for (int et = 0; et < 2; ++et) {
        const int e0 = wave * 32 + et * 16;

        // B-matrix (4x16 f32): lanes 0-15 -> N cols, VGPR0 = K0; K=1..3 zero.
        v2f Bm;
        Bm.x = (half == 0) ? w[e0 + l16] : 0.0f;
        Bm.y = 0.0f;

        // C-matrix (16x16 f32): VGPR j = row j / j+8, lane = column.
        // Bias depends only on the column -> replicate across all 8 VGPRs.
        const float bv = bias[e0 + l16];
        v8f C;
#pragma unroll
        for (int j = 0; j < 8; ++j) C[j] = bv;

        // D = A x B + C   (EXEC is all-1s: no divergent control flow above)
        v8f Dm = __builtin_amdgcn_wmma_f32_16x16x4_f32(
            /*neg_a=*/false, A, /*neg_b=*/false, Bm,
            /*c_mod=*/(short)0, C, /*reuse_a=*/false, /*reuse_b=*/false);

        // Store: VGPR j -> row t0 + j + 8*half, column e0 + l16.
        // Each b32 store = two contiguous 64B runs (lanes 0-15 / 16-31): coalesced.
        float* __restrict__ obase =
            out + ((size_t)bIdx * T_ + (size_t)(t0 + half * 8)) * E_ + (e0 + l16);
#pragma unroll
        for (int j = 0; j < 8; ++j) {
            obase[(size_t)j * E_] = Dm[j];
        }
    }
}

extern "C" void kernel_launch(void* const* d_in, const int* in_sizes, int n_in,
                              void* d_out, int out_size, void* d_ws, size_t ws_size,
                              hipStream_t stream) {
    const float* x          = (const float*)d_in[0];   // [B, T, D] f32
    const int*   asset_dims = (const int*)d_in[1];     // [B] int32
    const float* w          = (const float*)d_in[2];   // [E] f32
    const float* b          = (const float*)d_in[3];   // [E] f32
    float*       out        = (float*)d_out;           // [B, T, E] f32

    (void)in_sizes; (void)n_in; (void)out_size; (void)d_ws; (void)ws_size;

    const int blocks = B_ * (T_ / 16);                 // 8192 blocks, 8 waves each
    masked_proj_wmma_kernel<<<blocks, 256, 0, stream>>>(x, asset_dims, w, b, out);
}